// Hybrid_router_82197084111014
// MI455X (gfx1250) — compile-verified
//
#include <hip/hip_runtime.h>
#include <math.h>

typedef __attribute__((ext_vector_type(2))) float v2f;
typedef __attribute__((ext_vector_type(8))) float v8f;

#define NTOK   16384
#define CIN    2048
#define NE     16
#define ND     128
#define NA     128
#define NR     4
#define LN_EPS 1e-5f

// workspace layout (float offsets)
#define WS_WTP   0        // 65536 : combined router weight, WMMA-B layout: [(k/2)][n(32)][2]
#define WS_T     65536    // 1024  : T[e1][e2][r]
#define WS_U     66560    // 256
#define WS_V     66816    // 256
#define WS_WC    67072    // 256
#define WS_G     67328    // 16    : G[r][r'] = sum_o w2[o][r] w2[o][r']
#define WS_G2    67344    // 4     : sum_o w2[o][r] b2[o]
#define WS_CSUM  67348    // 4     : sum_o w2[o][r]
#define WS_SC    67352    // 2     : [c2 = sum b2^2, bsum = sum b2]

// ---------------------------------------------------------------------------
// Precompute kernel: collapses dyn->LN->q->sim chain into tiny tensors.
// ---------------------------------------------------------------------------
__global__ __launch_bounds__(256) void router_pre(
    const float* __restrict__ ee,      // (16,128) expert_embedding
    const float* __restrict__ gam,     // (16,128) ln_gamma
    const float* __restrict__ bet,     // (16,128) ln_beta
    const float* __restrict__ Wv_w,    // (16,2048)
    const float* __restrict__ mlp_w1,  // (4,2048)
    const float* __restrict__ w2,      // (2048,4)
    const float* __restrict__ b2,      // (2048,)
    const float* __restrict__ Wq,      // (128,128)
    const float* __restrict__ Wk,      // (128,128)
    float* __restrict__ ws)
{
  __shared__ float ge[NE * ND];
  __shared__ float kmat[NE * NA];
  __shared__ float Mm[ND * NE];
  __shared__ float red[256];
  const int tid = threadIdx.x;

  // ---- ge = LayerNorm(expert_embedding) over all 2048 elems ----
  float s = 0.f;
  for (int i = tid; i < NE * ND; i += 256) s += ee[i];
  red[tid] = s; __syncthreads();
  for (int off = 128; off > 0; off >>= 1) {
    if (tid < off) red[tid] += red[tid + off];
    __syncthreads();
  }
  const float mu = red[0] * (1.f / 2048.f);
  __syncthreads();
  float s2 = 0.f;
  for (int i = tid; i < NE * ND; i += 256) { float d = ee[i] - mu; s2 += d * d; }
  red[tid] = s2; __syncthreads();
  for (int off = 128; off > 0; off >>= 1) {
    if (tid < off) red[tid] += red[tid + off];
    __syncthreads();
  }
  const float rsig = rsqrtf(red[0] * (1.f / 2048.f) + LN_EPS);
  __syncthreads();
  for (int i = tid; i < NE * ND; i += 256)
    ge[i] = (ee[i] - mu) * rsig * gam[i] + bet[i];
  __syncthreads();

  // ---- kmat[e][a] = sum_d ge[e][d] * Wk[a][d] ----
  for (int idx = tid; idx < NE * NA; idx += 256) {
    const int e = idx >> 7, a = idx & 127;
    float acc = 0.f;
    for (int d = 0; d < ND; ++d) acc += ge[e * ND + d] * Wk[a * ND + d];
    kmat[idx] = acc;
  }
  __syncthreads();

  // ---- M[d][e2] = scale * sum_a Wq[a][d] * kmat[e2][a] ----
  const float scale = 0.08838834764831845f;  // 1/sqrt(128)
  for (int idx = tid; idx < ND * NE; idx += 256) {
    const int d = idx >> 4, e2 = idx & 15;
    float acc = 0.f;
    for (int a = 0; a < NA; ++a) acc += Wq[a * ND + d] * kmat[e2 * NA + a];
    Mm[d * NE + e2] = acc * scale;
  }
  __syncthreads();

  // ---- T[e1][e2][r] = sum_d w2[e1*128+d][r] * gamma[e1][d] * M[d][e2] ----
  for (int idx = tid; idx < 1024; idx += 256) {
    const int e1 = idx >> 6, e2 = (idx >> 2) & 15, r = idx & 3;
    float acc = 0.f;
    for (int d = 0; d < ND; ++d)
      acc += w2[(e1 * ND + d) * NR + r] * gam[e1 * ND + d] * Mm[d * NE + e2];
    ws[WS_T + idx] = acc;
  }
  // ---- U, V, Wc ----
  for (int idx = tid; idx < 256; idx += 256) {
    const int e1 = idx >> 4, e2 = idx & 15;
    float au = 0.f, av = 0.f, aw = 0.f;
    for (int d = 0; d < ND; ++d) {
      const float m = Mm[d * NE + e2];
      const float g = gam[e1 * ND + d];
      au += b2[e1 * ND + d] * g * m;
      av += g * m;
      aw += bet[e1 * ND + d] * m;
    }
    ws[WS_U + idx] = au; ws[WS_V + idx] = av; ws[WS_WC + idx] = aw;
  }
  // ---- quadratic-form LN stats of dyn ----
  if (tid < 16) {
    const int r = tid >> 2, rp = tid & 3;
    float acc = 0.f;
    for (int o = 0; o < CIN; ++o) acc += w2[o * NR + r] * w2[o * NR + rp];
    ws[WS_G + tid] = acc;
  } else if (tid < 20) {
    const int r = tid - 16;
    float a2 = 0.f, cs = 0.f;
    for (int o = 0; o < CIN; ++o) { const float w = w2[o * NR + r]; a2 += w * b2[o]; cs += w; }
    ws[WS_G2 + r] = a2; ws[WS_CSUM + r] = cs;
  } else if (tid == 20) {
    float cc = 0.f, bs = 0.f;
    for (int o = 0; o < CIN; ++o) { cc += b2[o] * b2[o]; bs += b2[o]; }
    ws[WS_SC + 0] = cc; ws[WS_SC + 1] = bs;
  }

  // ---- WTp: combined [Wv_w;mlp_w1;pad] transposed into f32 WMMA-B layout ----
  // flat index = ((k>>1)*32 + n)*2 + (k&1)
  for (int idx = tid; idx < 65536; idx += 256) {
    const int p = idx >> 6, n = (idx >> 1) & 31, j = idx & 1;
    const int k = 2 * p + j;
    float v = 0.f;
    if (n < 16)       v = Wv_w[n * CIN + k];
    else if (n < 20)  v = mlp_w1[(n - 16) * CIN + k];
    ws[WS_WTP + idx] = v;
  }
}

// ---------------------------------------------------------------------------
// Main kernel: one wave per 16-token tile. Phase A: fp32 WMMA GEMV over
// hidden (the single streaming pass). Phase B: collapsed routing math.
// ---------------------------------------------------------------------------
__global__ __launch_bounds__(256) void router_main(
    const float* __restrict__ hidden,  // (16384, 2048)
    const float* __restrict__ Wv_b,    // (16,)
    const float* __restrict__ b1,      // (4,)
    const float* __restrict__ ws,
    float* __restrict__ out)           // (16384, 16)
{
  __shared__ float sT[1024], sU[256], sV[256], sWc[256];
  __shared__ float sG[16], sg2[4], scs[4], ssc[2];
  __shared__ float sIG[8][16][17];  // per-wave ig[token][expert], padded
  __shared__ float sH[8][16][4];    // per-wave h[token][r]

  const int tid = threadIdx.x;
  for (int i = tid; i < 1024; i += 256) sT[i] = ws[WS_T + i];
  for (int i = tid; i < 256; i += 256) {
    sU[i] = ws[WS_U + i]; sV[i] = ws[WS_V + i]; sWc[i] = ws[WS_WC + i];
  }
  if (tid < 16) sG[tid] = ws[WS_G + tid];
  if (tid < 4) { sg2[tid] = ws[WS_G2 + tid]; scs[tid] = ws[WS_CSUM + tid]; }
  if (tid < 2) ssc[tid] = ws[WS_SC + tid];
  __syncthreads();

  const int wave = tid >> 5;
  const int lane = tid & 31;
  const int tile = blockIdx.x * 8 + wave;
  const int tokBase = tile * 16;

  const int mrow  = lane & 15;   // token row (A); N column (B/C)
  const int khalf = lane >> 4;   // 0: K=k0,k0+1  1: K=k0+2,k0+3

  const float* __restrict__ hrow = hidden + (size_t)(tokBase + mrow) * CIN + 2 * khalf;
  const float* __restrict__ bptr = ws + WS_WTP + (size_t)khalf * 64 + mrow * 2;

  v8f c0 = {};  // ig_pre  (experts 0..15)
  v8f c1 = {};  // mlp pre (cols 0..3 valid)
  #pragma unroll 4
  for (int k0 = 0; k0 < CIN; k0 += 4) {
    const v2f a   = *(const v2f*)(hrow + k0);
    const v2f bA  = *(const v2f*)(bptr + (size_t)k0 * 32);       // N-tile 0
    const v2f bB  = *(const v2f*)(bptr + (size_t)k0 * 32 + 32);  // N-tile 1
    c0 = __builtin_amdgcn_wmma_f32_16x16x4_f32(false, a, false, bA, 0, c0, false, false);
    c1 = __builtin_amdgcn_wmma_f32_16x16x4_f32(false, a, false, bB, 0, c1, false, false);
  }

  // scatter C-fragments (M = j + 8*khalf, N = mrow) into LDS
  #pragma unroll
  for (int j = 0; j < 8; ++j) {
    const int t = j + 8 * khalf;
    sIG[wave][t][mrow] = c0[j] + Wv_b[mrow];
    if (mrow < 4) {
      const float hv = c1[j] + b1[mrow];
      sH[wave][t][mrow] = hv > 0.f ? hv : 0.f;
    }
  }
  __syncthreads();

  // ---- Phase B: lane t < 16 handles token t of this wave's tile ----
  if (lane < 16) {
    const int t = lane;
    float h[4];
    #pragma unroll
    for (int r = 0; r < 4; ++r) h[r] = sH[wave][t][r];

    // LN stats of dyn via quadratic form
    float S1 = ssc[1];
    #pragma unroll
    for (int r = 0; r < 4; ++r) S1 += h[r] * scs[r];
    const float mu = S1 * (1.f / 2048.f);
    float S2 = ssc[0];
    #pragma unroll
    for (int r = 0; r < 4; ++r) {
      S2 += 2.f * h[r] * sg2[r];
      #pragma unroll
      for (int rp = 0; rp < 4; ++rp) S2 += h[r] * h[rp] * sG[r * 4 + rp];
    }
    const float var  = S2 * (1.f / 2048.f) - mu * mu;
    const float rsig = rsqrtf(var + LN_EPS);

    float ig[NE];
    #pragma unroll
    for (int e = 0; e < NE; ++e) ig[e] = sIG[wave][t][e];

    // gate[e1] = softmax_e2(sim_pre[e1][e2]) . ig
    float gate[NE];
    #pragma unroll
    for (int e1 = 0; e1 < NE; ++e1) {
      float sp[NE];
      float mx = -3.4e38f;
      #pragma unroll
      for (int e2 = 0; e2 < NE; ++e2) {
        const int q = e1 * 16 + e2;
        float v = sU[q] - mu * sV[q];
        v += h[0] * sT[q * 4 + 0] + h[1] * sT[q * 4 + 1] +
             h[2] * sT[q * 4 + 2] + h[3] * sT[q * 4 + 3];
        v = rsig * v + sWc[q];
        sp[e2] = v;
        mx = fmaxf(mx, v);
      }
      float den = 0.f, acc = 0.f;
      #pragma unroll
      for (int e2 = 0; e2 < NE; ++e2) {
        const float e = __expf(sp[e2] - mx);
        den += e;
        acc += e * ig[e2];
      }
      gate[e1] = acc / den;
    }

    // rw = softmax(gate)
    float m1 = -3.4e38f;
    #pragma unroll
    for (int e = 0; e < NE; ++e) m1 = fmaxf(m1, gate[e]);
    float rw[NE]; float den = 0.f;
    #pragma unroll
    for (int e = 0; e < NE; ++e) { rw[e] = __expf(gate[e] - m1); den += rw[e]; }
    const float rden = 1.f / den;
    #pragma unroll
    for (int e = 0; e < NE; ++e) rw[e] *= rden;

    // top-2 (first occurrence wins ties, matching lax.top_k)
    float v1 = -1.f; int i1 = 0;
    #pragma unroll
    for (int e = 0; e < NE; ++e) if (rw[e] > v1) { v1 = rw[e]; i1 = e; }
    float v2 = -1.f; int i2 = 0;
    #pragma unroll
    for (int e = 0; e < NE; ++e) if (e != i1 && rw[e] > v2) { v2 = rw[e]; i2 = e; }

    // sharpened softmax over {v1, v2}; all masked entries underflow to 0
    const float p2  = __expf((v2 - v1) * 10.f);
    const float inv = 1.f / (1.f + p2);

    float* __restrict__ orow = out + (size_t)(tokBase + t) * NE;
    #pragma unroll
    for (int e = 0; e < NE; ++e) {
      const float o = (e == i1) ? inv : ((e == i2) ? p2 * inv : 0.f);
      orow[e] = o;
    }
  }
}

// ---------------------------------------------------------------------------
extern "C" void kernel_launch(void* const* d_in, const int* in_sizes, int n_in,
                              void* d_out, int out_size, void* d_ws, size_t ws_size,
                              hipStream_t stream) {
  const float* hidden  = (const float*)d_in[0];   // (4,4096,2048)
  const float* ee      = (const float*)d_in[1];   // (16,128)
  const float* gam     = (const float*)d_in[2];
  const float* bet     = (const float*)d_in[3];
  const float* Wv_w    = (const float*)d_in[4];   // (16,2048)
  const float* Wv_b    = (const float*)d_in[5];   // (16,)
  const float* mlp_w1  = (const float*)d_in[6];   // (4,2048)
  const float* mlp_b1  = (const float*)d_in[7];   // (4,)
  const float* mlp_w2  = (const float*)d_in[8];   // (2048,4)
  const float* mlp_b2  = (const float*)d_in[9];   // (2048,)
  const float* te_Wq   = (const float*)d_in[10];  // (128,128)
  const float* te_Wk   = (const float*)d_in[11];  // (128,128)
  // d_in[12] = top_k (always 2 in reference setup)

  float* ws  = (float*)d_ws;
  float* o   = (float*)d_out;

  router_pre<<<1, 256, 0, stream>>>(ee, gam, bet, Wv_w, mlp_w1, mlp_w2, mlp_b2,
                                    te_Wq, te_Wk, ws);
  router_main<<<128, 256, 0, stream>>>(hidden, Wv_b, mlp_b1, ws, o);
}